// RHN_34162169872947
// MI455X (gfx1250) — compile-verified
//
#include <hip/hip_runtime.h>

// RHN (recurrent highway net) persistent-scan kernel for gfx1250.
// Batch rows are independent through the recurrence -> 4 persistent
// workgroups, each owning a 16-row (WMMA-M) batch tile for all 2048 steps.
// fp32 WMMA (v_wmma_f32_16x16x4_f32) keeps reference precision.
// LDS tiles use a 260-float row pitch so WMMA A-operand ds_load_b64s are
// 64-bank conflict-free. The hidden state is carried in registers (each lane
// owns the same 8 D-matrix elements it produces), LDS holds it only as the
// WMMA A-operand source; biases are hoisted into registers before the scan.

typedef __attribute__((ext_vector_type(2))) float v2f;
typedef __attribute__((ext_vector_type(4))) float v4f;
typedef __attribute__((ext_vector_type(8))) float v8f;

#define SEQ    2048
#define BATCH  64
#define HID    256
#define HPAD   260        // LDS row pitch (floats): 260 mod 64 == 4 -> no conflicts
#define NSTEP  3
#define KSTEPS (HID / 4)  // 64 WMMA k-steps per K=256 GEMM

__global__ __launch_bounds__(512, 1)
void rhn_persistent_kernel(const float* __restrict__ x,    // [SEQ][BATCH][HID]
                           const float* __restrict__ h0,   // [1][BATCH][HID]
                           const float* __restrict__ Wi,   // [HID][HID] (in,out)
                           const float* __restrict__ bi,   // [HID]
                           const float* __restrict__ Hw,   // [NSTEP][HID][HID]
                           const float* __restrict__ Hb,   // [NSTEP][HID]
                           float* __restrict__ out)        // [SEQ][BATCH][HID]
{
    __shared__ float hsh[16 * HPAD];       // hidden state tile (A-operand source)
    __shared__ float xsh[16 * HPAD];       // staged x tile for this timestep

    const int tid  = threadIdx.x;
    const int wave = tid >> 5;             // 16 waves -> 16 n-tiles (16 cols each)
    const int lane = tid & 31;
    const int half = lane >> 4;            // lane half selects k/row sub-range
    const int idx  = lane & 15;
    const int b0   = blockIdx.x << 4;      // batch row offset of this tile
    const int n    = (wave << 4) + idx;    // global hidden column owned by lane

    // Stage initial hidden state into LDS (A-operand layout).
    for (int i = tid; i < 16 * HID; i += 512) {
        int r = i >> 8, c = i & 255;
        hsh[r * HPAD + c] = h0[b0 * HID + i];
    }

    // Per-lane persistent state: the 8 D-matrix elements (rows r+8*half, col n)
    // this lane produces each microstep.  Identical mapping every microstep,
    // so h_old for the highway combine never needs to be re-read from LDS.
    float hreg[8];
    #pragma unroll
    for (int r = 0; r < 8; ++r)
        hreg[r] = h0[(b0 + r + 8 * half) * HID + n];

    // Time-invariant biases, hoisted out of the scan (bi folded into step 0).
    float bias_s[NSTEP];
    bias_s[0] = Hb[n] + bi[n];
    #pragma unroll
    for (int s = 1; s < NSTEP; ++s) bias_s[s] = Hb[s * HID + n];

    __syncthreads();

    // A-operand base: lane supplies h[m = idx][k = 4j + 2*half .. +1] (b64 LDS load)
    const int arow = idx * HPAD + 2 * half;

    for (int t = 0; t < SEQ; ++t) {
        // Stage x tile (16x256 -> padded pitch) with coalesced b128 loads.
        {
            const float* xt = x + (size_t)t * (BATCH * HID) + (size_t)b0 * HID;
            #pragma unroll
            for (int f = tid * 4; f < 16 * HID; f += 512 * 4) {
                int r = f >> 8, c = f & 255;
                *(v4f*)&xsh[r * HPAD + c] = *(const v4f*)&xt[f];
            }
        }
        // Prefetch next timestep's x tile so HBM latency overlaps the 3
        // microsteps of compute (global_prefetch_b8).
        if (t + 1 < SEQ) {
            const float* xn = x + (size_t)(t + 1) * (BATCH * HID)
                                + (size_t)b0 * HID;
            __builtin_prefetch(xn + tid * 8, 0, 1);
        }
        __syncthreads();

        for (int s = 0; s < NSTEP; ++s) {
            v8f acc = {};
            // Recurrent GEMM: h(16x256) @ Hw[s](256x256), this wave's 16 cols.
            const float* wp = Hw + (size_t)s * (HID * HID) + (2 * half) * HID + n;
            #pragma unroll 8
            for (int j = 0; j < KSTEPS; ++j) {
                v2f a = *(const v2f*)(hsh + arow + 4 * j);
                v2f b;
                b.x = wp[4 * j * HID];          // W[k  ][n]
                b.y = wp[4 * j * HID + HID];    // W[k+1][n]
                acc = __builtin_amdgcn_wmma_f32_16x16x4_f32(
                          false, a, false, b, (short)0, acc, false, false);
            }
            if (s == 0) {
                // Fused input projection: x_t(16x256) @ Wi(256x256).
                const float* wq = Wi + (2 * half) * HID + n;
                #pragma unroll 8
                for (int j = 0; j < KSTEPS; ++j) {
                    v2f a = *(const v2f*)(xsh + arow + 4 * j);
                    v2f b;
                    b.x = wq[4 * j * HID];
                    b.y = wq[4 * j * HID + HID];
                    acc = __builtin_amdgcn_wmma_f32_16x16x4_f32(
                              false, a, false, b, (short)0, acc, false, false);
                }
            }
            // Highway combine: h = tanh(a)*sig(a) + h*(1 - sig(a)).
            const float biasv = bias_s[s];
            #pragma unroll
            for (int r = 0; r < 8; ++r) {
                float av = acc[r] + biasv;
                float e2 = __expf(-2.f * av);
                float th = (1.f - e2) / (1.f + e2);    // tanh(av)
                float e1 = __expf(-av);
                float sg = 1.f / (1.f + e1);           // sigmoid(av)
                hreg[r] = th * sg + hreg[r] * (1.f - sg);
            }
            __syncthreads();   // every wave done reading hsh for this microstep
            #pragma unroll
            for (int r = 0; r < 8; ++r)
                hsh[(r + 8 * half) * HPAD + n] = hreg[r];
            __syncthreads();   // hsh updated for next microstep
        }
        // Emit this timestep's hidden state from registers.
        float* ot = out + (size_t)t * (BATCH * HID) + b0 * HID;
        #pragma unroll
        for (int r = 0; r < 8; ++r)
            ot[(r + 8 * half) * HID + n] = hreg[r];
    }
}

extern "C" void kernel_launch(void* const* d_in, const int* in_sizes, int n_in,
                              void* d_out, int out_size, void* d_ws, size_t ws_size,
                              hipStream_t stream) {
    (void)in_sizes; (void)n_in; (void)out_size; (void)d_ws; (void)ws_size;
    const float* x  = (const float*)d_in[0];   // [2048,64,256]
    const float* h0 = (const float*)d_in[1];   // [1,64,256]
    const float* Wi = (const float*)d_in[2];   // [256,256]
    const float* bi = (const float*)d_in[3];   // [256]
    const float* Hw = (const float*)d_in[4];   // [3,256,256]
    const float* Hb = (const float*)d_in[5];   // [3,256]
    float* out = (float*)d_out;                // [2048,64,256]

    dim3 grid(BATCH / 16);   // 4 persistent workgroups (independent batch tiles)
    dim3 block(512);         // 16 wave32 -> one n-tile of 16 columns per wave
    rhn_persistent_kernel<<<grid, block, 0, stream>>>(x, h0, Wi, bi, Hw, Hb, out);
}